// critic_10582799417477
// MI455X (gfx1250) — compile-verified
//
#include <hip/hip_runtime.h>

// ---------------------------------------------------------------------------
// GCN critic on MI455X (gfx1250, wave32).
//  - All dense math on V_WMMA_F32_16X16X4_F32 (fp32 in/acc, exact precision).
//  - Dominant op (emb matmul: 327 MB streamed, ~32 FLOP/B -> HBM bound at
//    23.3 TB/s, floor ~14us) streams A with nontemporal loads while the
//    L2-resident B panel is staged through LDS with CDNA5 async-to-LDS DMA
//    (global_load_async_to_lds_b128 + s_wait_asynccnt), double buffered.
// ---------------------------------------------------------------------------

typedef __attribute__((ext_vector_type(2))) float v2f;
typedef __attribute__((ext_vector_type(8))) float v8f;

#define BB   4
#define NLOG 10000
#define NHW  2048
#define ELOG 160000
#define EHW  2048

static __device__ __forceinline__ void atomic_add_f32(float* p, float v) {
  __hip_atomic_fetch_add(p, v, __ATOMIC_RELAXED, __HIP_MEMORY_SCOPE_AGENT);
}

// ---------------- small utility kernels ----------------

__global__ void fill_kernel(float* __restrict__ p, float v, int count) {
  int t = blockIdx.x * blockDim.x + threadIdx.x;
  if (t < count) p[t] = v;
}

// deg[b*n + dst] += 1 for every edge (self-loop "1" pre-seeded by fill)
__global__ void deg_count_kernel(const int* __restrict__ edges,
                                 float* __restrict__ deg, int E, int n) {
  int t = blockIdx.x * blockDim.x + threadIdx.x;
  if (t >= BB * E) return;
  int b = t / E;
  int e = t - b * E;
  int dst = edges[(size_t)b * 2 * E + E + e];
  atomic_add_f32(deg + (size_t)b * n + dst, 1.0f);
}

__global__ void rsqrt_kernel(float* __restrict__ p, int count) {
  int t = blockIdx.x * blockDim.x + threadIdx.x;
  if (t < count) p[t] = rsqrtf(p[t]);  // deg >= 1 always (self loops)
}

// ---------------- WMMA fp32 GEMM: H[b] = X[b] (n x k) @ W (k x 64) ----------
// One wave computes a 16x64 tile: 4 accumulators of 16x16, K-loop step 4.
// A layout (16x4 f32): lane = 16*(K>>1) + M, vgpr = K&1
// B layout (4x16 f32): lane = 16*(K>>1) + N, vgpr = K&1
// D layout (16x16 f32): vgpr v -> M = v + 8*(lane>>4), N = lane&15
__global__ void gemm16_wmma(const float* __restrict__ X,
                            const float* __restrict__ W,
                            float* __restrict__ H, int n, int k,
                            int totalTiles) {
  int tile = blockIdx.x * (blockDim.x >> 5) + (threadIdx.x >> 5);
  if (tile >= totalTiles) return;  // wave-uniform; EXEC stays all-1s
  int tilesPer = n >> 4;
  int b = tile / tilesPer;
  int m0 = (tile - b * tilesPer) << 4;

  int lane = threadIdx.x & 31;
  int half = lane >> 4;   // 0/1 -> K pair {0,1} vs {2,3}
  int idx  = lane & 15;   // M for A, N for B

  const float* xb = X + (size_t)b * n * k;
  float* hb = H + ((size_t)b * n + m0) * 64;

  v8f acc0 = {}, acc1 = {}, acc2 = {}, acc3 = {};
  const float* arow = xb + (size_t)(m0 + idx) * k + 2 * half;

  for (int k0 = 0; k0 < k; k0 += 4) {
    v2f a = *(const v2f*)(arow + k0);
    const float* wp = W + (size_t)(k0 + 2 * half) * 64 + idx;
    v2f b0 = { wp[0],  wp[64]  };
    v2f b1 = { wp[16], wp[80]  };
    v2f b2 = { wp[32], wp[96]  };
    v2f b3 = { wp[48], wp[112] };
    acc0 = __builtin_amdgcn_wmma_f32_16x16x4_f32(false, a, false, b0,
                                                 (short)0, acc0, false, false);
    acc1 = __builtin_amdgcn_wmma_f32_16x16x4_f32(false, a, false, b1,
                                                 (short)0, acc1, false, false);
    acc2 = __builtin_amdgcn_wmma_f32_16x16x4_f32(false, a, false, b2,
                                                 (short)0, acc2, false, false);
    acc3 = __builtin_amdgcn_wmma_f32_16x16x4_f32(false, a, false, b3,
                                                 (short)0, acc3, false, false);
  }

#pragma unroll
  for (int v = 0; v < 8; v++) {
    float* o = hb + (size_t)(v + 8 * half) * 64 + idx;
    o[0]  = acc0[v];
    o[16] = acc1[v];
    o[32] = acc2[v];
    o[48] = acc3[v];
  }
}

// ---------------- emb matmul with async-LDS staged B panel ------------------
// OUT[b] = EMB[b] (NL x NH) @ XH[b] (NH x 64)
// 160-thread blocks (5 waves), each wave owns a 16x64 tile; the 5 waves share
// one batch's B panel. B chunks (KC x 64 fp32 = 32KB) are DMA'd into LDS via
// global_load_async_to_lds_b128 (ASYNCcnt), double buffered.
#define KC 128
#define EMB_CHUNKS (NHW / KC)  // 16
#define EMB_TPB 160

static __device__ __forceinline__ void async_copy_chunk(unsigned lds_base,
                                                        const float* __restrict__ src,
                                                        int tid) {
  // KC*64 floats = KC*16 float4s, linear memcpy global -> LDS
  for (int j = tid; j < KC * 16; j += EMB_TPB) {
    unsigned laddr = lds_base + (unsigned)(j * 16);
    unsigned long long gaddr = (unsigned long long)(const void*)(src + j * 4);
    asm volatile("global_load_async_to_lds_b128 %0, %1, off"
                 :: "v"(laddr), "v"(gaddr)
                 : "memory");
  }
}

__global__ void __launch_bounds__(EMB_TPB)
emb_gemm_wmma_lds(const float* __restrict__ EMB, const float* __restrict__ XH,
                  float* __restrict__ OUT) {
  __shared__ float smem[2][KC * 64];  // 2 x 32KB

  int b = blockIdx.y;
  int tid = threadIdx.x;
  int wave = tid >> 5;
  int lane = tid & 31;
  int half = lane >> 4;
  int idx  = lane & 15;
  int m0 = blockIdx.x * 80 + wave * 16;  // 125*80 == NLOG exactly

  const float* xb  = EMB + ((size_t)b * NLOG + m0) * NHW;
  const float* xhb = XH + (size_t)b * NHW * 64;
  float* ob = OUT + ((size_t)b * NLOG + m0) * 64;

  const float* arow = xb + (size_t)idx * NHW + 2 * half;

  unsigned lds0 = (unsigned)(size_t)(void*)&smem[0][0];
  unsigned lds1 = (unsigned)(size_t)(void*)&smem[1][0];

  v8f acc0 = {}, acc1 = {}, acc2 = {}, acc3 = {};

  // prime chunk 0
  async_copy_chunk(lds0, xhb, tid);
  asm volatile("s_wait_asynccnt 0x0" ::: "memory");
  __syncthreads();

  for (int c = 0; c < EMB_CHUNKS; c++) {
    int cur = c & 1;
    if (c + 1 < EMB_CHUNKS)  // prefetch next chunk into the other buffer
      async_copy_chunk(cur ? lds0 : lds1, xhb + (size_t)(c + 1) * KC * 64, tid);

    const float* bs = &smem[cur][0];
    const float* ap = arow + c * KC;
    for (int kc = 0; kc < KC; kc += 4) {
      // A (emb) is streamed once and > L2 capacity -> nontemporal
      v2f a = __builtin_nontemporal_load((const v2f*)(ap + kc));
      const float* wp = bs + (size_t)(kc + 2 * half) * 64 + idx;
      v2f b0 = { wp[0],  wp[64]  };
      v2f b1 = { wp[16], wp[80]  };
      v2f b2 = { wp[32], wp[96]  };
      v2f b3 = { wp[48], wp[112] };
      acc0 = __builtin_amdgcn_wmma_f32_16x16x4_f32(false, a, false, b0,
                                                   (short)0, acc0, false, false);
      acc1 = __builtin_amdgcn_wmma_f32_16x16x4_f32(false, a, false, b1,
                                                   (short)0, acc1, false, false);
      acc2 = __builtin_amdgcn_wmma_f32_16x16x4_f32(false, a, false, b2,
                                                   (short)0, acc2, false, false);
      acc3 = __builtin_amdgcn_wmma_f32_16x16x4_f32(false, a, false, b3,
                                                   (short)0, acc3, false, false);
    }

    // my async writes for chunk c+1 done; barrier also guarantees every wave
    // finished reading buffer `cur` before it is overwritten next iteration
    asm volatile("s_wait_asynccnt 0x0" ::: "memory");
    __syncthreads();
  }

#pragma unroll
  for (int v = 0; v < 8; v++) {
    float* o = ob + (size_t)(v + 8 * half) * 64 + idx;
    o[0]  = acc0[v];
    o[16] = acc1[v];
    o[32] = acc2[v];
    o[48] = acc3[v];
  }
}

// ---------------- GCN aggregation ----------------
// out = bias + h * dinv^2   (self-loop term), layout [B, n, 64]
__global__ void agg_init_kernel(const float* __restrict__ h,
                                const float* __restrict__ dinv,
                                const float* __restrict__ bias,
                                float* __restrict__ out, long long total) {
  long long t = (long long)blockIdx.x * blockDim.x + threadIdx.x;
  if (t >= total) return;
  int c = (int)(t & 63);
  long long bi = t >> 6;  // b*n + i
  float d = dinv[bi];
  out[t] = bias[c] + h[t] * d * d;
}

// out[dst] += h[src] * dinv[src]*dinv[dst], one thread per (edge, channel)
__global__ void agg_scatter_kernel(const float* __restrict__ h,
                                   const float* __restrict__ dinv,
                                   const int* __restrict__ edges,
                                   float* __restrict__ out, int n, int E) {
  long long t = (long long)blockIdx.x * blockDim.x + threadIdx.x;
  long long total = (long long)BB * E * 64;
  if (t >= total) return;
  int c = (int)(t & 63);
  long long r = t >> 6;  // b*E + e
  int b = (int)(r / E);
  int e = (int)(r - (long long)b * E);
  const int* eb = edges + (size_t)b * 2 * E;
  int src = eb[e];
  int dst = eb[E + e];
  size_t bs = (size_t)b * n + src;
  size_t bd = (size_t)b * n + dst;
  float w = dinv[bs] * dinv[bd];
  atomic_add_f32(out + bd * 64 + c, h[bs * 64 + c] * w);
}

// ---------------- final fused head ----------------

__global__ void out_init_kernel(const float* __restrict__ lin2b,
                                float* __restrict__ out) {
  int t = threadIdx.x;
  if (t < BB) out[t] = lin2b[0];
}

// out[b] += sum_i ( dot(concat(xl[i], xe[i]), linW) + linb ) * lin2W[i]
__global__ void final_reduce_kernel(const float* __restrict__ xl,
                                    const float* __restrict__ xe,
                                    const float* __restrict__ linW,
                                    const float* __restrict__ linb,
                                    const float* __restrict__ lin2W,
                                    float* __restrict__ out) {
  int b = blockIdx.y;
  int i = blockIdx.x * blockDim.x + threadIdx.x;
  float val = 0.0f;
  if (i < NLOG) {
    const float4* pl = (const float4*)(xl + ((size_t)b * NLOG + i) * 64);
    const float4* pe = (const float4*)(xe + ((size_t)b * NLOG + i) * 64);
    const float4* w  = (const float4*)linW;
    float s = linb[0];
#pragma unroll
    for (int q = 0; q < 16; q++) {
      float4 a = pl[q], ww = w[q];
      s += a.x * ww.x + a.y * ww.y + a.z * ww.z + a.w * ww.w;
    }
#pragma unroll
    for (int q = 0; q < 16; q++) {
      float4 a = pe[q], ww = w[16 + q];
      s += a.x * ww.x + a.y * ww.y + a.z * ww.z + a.w * ww.w;
    }
    val = s * lin2W[i];
  }
  __shared__ float red[256];
  red[threadIdx.x] = val;
  __syncthreads();
  for (int s2 = 128; s2 > 0; s2 >>= 1) {
    if (threadIdx.x < s2) red[threadIdx.x] += red[threadIdx.x + s2];
    __syncthreads();
  }
  if (threadIdx.x == 0) atomic_add_f32(out + b, red[0]);
}

// ---------------------------------------------------------------------------

static inline int cdiv(long long a, long long b) { return (int)((a + b - 1) / b); }

extern "C" void kernel_launch(void* const* d_in, const int* in_sizes, int n_in,
                              void* d_out, int out_size, void* d_ws,
                              size_t ws_size, hipStream_t stream) {
  const float* x_l   = (const float*)d_in[0];
  const int*   e_l   = (const int*)d_in[1];
  const float* x_h   = (const float*)d_in[2];
  const int*   e_h   = (const int*)d_in[3];
  const float* emb   = (const float*)d_in[4];
  const float* Wl1   = (const float*)d_in[5];
  const float* bl1   = (const float*)d_in[6];
  const float* Wl2   = (const float*)d_in[7];
  const float* bl2   = (const float*)d_in[8];
  const float* Wl3   = (const float*)d_in[9];
  const float* bl3   = (const float*)d_in[10];
  const float* Wh1   = (const float*)d_in[11];
  const float* bh1   = (const float*)d_in[12];
  const float* Wh2   = (const float*)d_in[13];
  const float* bh2   = (const float*)d_in[14];
  const float* Wh3   = (const float*)d_in[15];
  const float* bh3   = (const float*)d_in[16];
  const float* linW  = (const float*)d_in[17];
  const float* linb  = (const float*)d_in[18];
  const float* lin2W = (const float*)d_in[19];
  const float* lin2b = (const float*)d_in[20];
  float* out = (float*)d_out;

  // workspace carve-up (floats); ~37 MB total
  float* ws = (float*)d_ws;
  size_t off = 0;
  float* dinv_l = ws + off; off += (size_t)BB * NLOG;
  float* dinv_h = ws + off; off += (size_t)BB * NHW;
  float* Hl = ws + off; off += (size_t)BB * NLOG * 64;
  float* Al = ws + off; off += (size_t)BB * NLOG * 64;
  float* Bl = ws + off; off += (size_t)BB * NLOG * 64;
  float* Hh = ws + off; off += (size_t)BB * NHW * 64;
  float* Ah = ws + off; off += (size_t)BB * NHW * 64;
  float* Bh = ws + off; off += (size_t)BB * NHW * 64;

  const int TPB = 256;

  // ---- degree -> dinv (shared across the 3 layers of each graph) ----
  fill_kernel<<<cdiv(BB * NLOG, TPB), TPB, 0, stream>>>(dinv_l, 1.0f, BB * NLOG);
  fill_kernel<<<cdiv(BB * NHW, TPB), TPB, 0, stream>>>(dinv_h, 1.0f, BB * NHW);
  deg_count_kernel<<<cdiv((long long)BB * ELOG, TPB), TPB, 0, stream>>>(e_l, dinv_l, ELOG, NLOG);
  deg_count_kernel<<<cdiv((long long)BB * EHW, TPB), TPB, 0, stream>>>(e_h, dinv_h, EHW, NHW);
  rsqrt_kernel<<<cdiv(BB * NLOG, TPB), TPB, 0, stream>>>(dinv_l, BB * NLOG);
  rsqrt_kernel<<<cdiv(BB * NHW, TPB), TPB, 0, stream>>>(dinv_h, BB * NHW);

  const int tilesL = BB * (NLOG / 16);  // 2500
  const int tilesH = BB * (NHW / 16);   // 512
  const long long aggL = (long long)BB * NLOG * 64;
  const long long aggH = (long long)BB * NHW * 64;
  const int scatLblocks = cdiv((long long)BB * ELOG * 64, TPB);
  const int scatHblocks = cdiv((long long)BB * EHW * 64, TPB);

  // ---- logical GCN stack (16->64->64->64) ----
  gemm16_wmma<<<cdiv(tilesL, 4), 128, 0, stream>>>(x_l, Wl1, Hl, NLOG, 16, tilesL);
  agg_init_kernel<<<cdiv(aggL, TPB), TPB, 0, stream>>>(Hl, dinv_l, bl1, Al, aggL);
  agg_scatter_kernel<<<scatLblocks, TPB, 0, stream>>>(Hl, dinv_l, e_l, Al, NLOG, ELOG);

  gemm16_wmma<<<cdiv(tilesL, 4), 128, 0, stream>>>(Al, Wl2, Hl, NLOG, 64, tilesL);
  agg_init_kernel<<<cdiv(aggL, TPB), TPB, 0, stream>>>(Hl, dinv_l, bl2, Bl, aggL);
  agg_scatter_kernel<<<scatLblocks, TPB, 0, stream>>>(Hl, dinv_l, e_l, Bl, NLOG, ELOG);

  gemm16_wmma<<<cdiv(tilesL, 4), 128, 0, stream>>>(Bl, Wl3, Hl, NLOG, 64, tilesL);
  agg_init_kernel<<<cdiv(aggL, TPB), TPB, 0, stream>>>(Hl, dinv_l, bl3, Al, aggL);
  agg_scatter_kernel<<<scatLblocks, TPB, 0, stream>>>(Hl, dinv_l, e_l, Al, NLOG, ELOG);
  // xl final = Al

  // ---- hardware GCN stack ----
  gemm16_wmma<<<cdiv(tilesH, 4), 128, 0, stream>>>(x_h, Wh1, Hh, NHW, 16, tilesH);
  agg_init_kernel<<<cdiv(aggH, TPB), TPB, 0, stream>>>(Hh, dinv_h, bh1, Ah, aggH);
  agg_scatter_kernel<<<scatHblocks, TPB, 0, stream>>>(Hh, dinv_h, e_h, Ah, NHW, EHW);

  gemm16_wmma<<<cdiv(tilesH, 4), 128, 0, stream>>>(Ah, Wh2, Hh, NHW, 64, tilesH);
  agg_init_kernel<<<cdiv(aggH, TPB), TPB, 0, stream>>>(Hh, dinv_h, bh2, Bh, aggH);
  agg_scatter_kernel<<<scatHblocks, TPB, 0, stream>>>(Hh, dinv_h, e_h, Bh, NHW, EHW);

  gemm16_wmma<<<cdiv(tilesH, 4), 128, 0, stream>>>(Bh, Wh3, Hh, NHW, 64, tilesH);
  agg_init_kernel<<<cdiv(aggH, TPB), TPB, 0, stream>>>(Hh, dinv_h, bh3, Ah, aggH);
  agg_scatter_kernel<<<scatHblocks, TPB, 0, stream>>>(Hh, dinv_h, e_h, Ah, NHW, EHW);
  // xh final = Ah

  // ---- x_emb[b] = emb[b] (NL x NH) @ xh[b] (NH x 64): HBM-bound, async-LDS B
  emb_gemm_wmma_lds<<<dim3(125, BB), EMB_TPB, 0, stream>>>(emb, Ah, Bl);
  // x_emb = Bl

  // ---- fused head: concat -> linW -> lin2W -> [B] ----
  out_init_kernel<<<1, 32, 0, stream>>>(lin2b, out);
  dim3 grid(cdiv(NLOG, 256), BB);
  final_reduce_kernel<<<grid, 256, 0, stream>>>(Al, Bl, linW, linb, lin2W, out);
}